// BaselineModel_81965155877638
// MI455X (gfx1250) — compile-verified
//
#include <hip/hip_runtime.h>
#include <hip/hip_bf16.h>

typedef __attribute__((ext_vector_type(2))) float v2f;
typedef __attribute__((ext_vector_type(8))) float v8f;

#define NUSER 200000
#define NREST 50000
#define NEDGE 1000000
#define NLBL  500000
#define DIM   64

// ---------------- workspace layout (floats) ----------------
// [agg_rest 50000*64][agg_user 200000*64][cnt_rest 50000][cnt_user 200000][h_rest 50000*64][h_user 200000*64]
#define OFF_AGG_REST 0
#define OFF_AGG_USER (NREST * DIM)                         // 3,200,000
#define OFF_CNT_REST (OFF_AGG_USER + NUSER * DIM)          // 16,000,000
#define OFF_CNT_USER (OFF_CNT_REST + NREST)                // 16,050,000
#define OFF_H_REST   (OFF_CNT_USER + NUSER)                // 16,250,000
#define OFF_H_USER   (OFF_H_REST + NREST * DIM)            // 19,450,000
#define WS_FLOATS    (OFF_H_USER + NUSER * DIM)            // 32,250,000

// ---------------- utility kernels ----------------
__global__ void zero_f32(float* p, long long n) {
  long long i = (long long)blockIdx.x * blockDim.x + threadIdx.x;
  if (i < n) p[i] = 0.0f;
}

__global__ void count_edges(const int* __restrict__ su, const int* __restrict__ du,
                            float* cnt_rest, float* cnt_user, int ne) {
  int e = blockIdx.x * blockDim.x + threadIdx.x;
  if (e >= ne) return;
  atomicAdd(&cnt_rest[du[e]], 1.0f);
  atomicAdd(&cnt_user[su[e]], 1.0f);
}

__global__ void invert_cnt(float* c, int n) {
  int i = blockIdx.x * blockDim.x + threadIdx.x;
  if (i < n) c[i] = 1.0f / fmaxf(c[i], 1.0f);
}

// One pass over edge list feeds BOTH aggregation directions.
// thread t: edge e = t>>6, feature f = t&63 (64 consecutive threads share an edge:
// fully coalesced row reads and row-contiguous atomics).
__global__ void scatter_both(const float* __restrict__ xu, const float* __restrict__ xr,
                             const int* __restrict__ su, const int* __restrict__ du,
                             float* agg_rest, float* agg_user, int ne) {
  long long t = (long long)blockIdx.x * blockDim.x + threadIdx.x;
  if (t >= (long long)ne * DIM) return;
  int e = (int)(t >> 6);
  int f = (int)(t & 63);
  int u = su[e];
  int r = du[e];
  atomicAdd(&agg_rest[r * DIM + f], xu[u * DIM + f]);
  atomicAdd(&agg_user[u * DIM + f], xr[r * DIM + f]);
}

// ---------------- WMMA SAGE kernel ----------------
// out[m,:] = (agg[m,:]*invcnt[m]) @ Wl + bias + xdst[m,:] @ Wr   (optional ReLU)
// One wave handles a 16-row tile, full N=64, K=64 -> 16 k-chunks x 4 n-tiles x 2 matmuls
// = 128 v_wmma_f32_16x16x4_f32 per wave. Weights + bias staged in LDS.
__global__ void sage_wmma(const float* agg, const float* __restrict__ invcnt,
                          const float* __restrict__ xdst,
                          const float* __restrict__ Wl, const float* __restrict__ bias,
                          const float* __restrict__ Wr,
                          float* out, int nrows, int do_relu) {
  __shared__ float lwl[DIM * DIM];
  __shared__ float lwr[DIM * DIM];
  __shared__ float lb[DIM];
  for (int i = threadIdx.x; i < DIM * DIM; i += blockDim.x) {
    lwl[i] = Wl[i];
    lwr[i] = Wr[i];
  }
  if (threadIdx.x < DIM) lb[threadIdx.x] = bias[threadIdx.x];
  __syncthreads();

  const int lane = threadIdx.x & 31;
  const int wid  = threadIdx.x >> 5;
  const int row0 = (blockIdx.x * 8 + wid) * 16;
  if (row0 >= nrows) return;

  const int m    = lane & 15;   // A row / D column index
  const int half = lane >> 4;   // 0: K pair {0,1}, 1: K pair {2,3}
  const int kb   = half * 2;

  const float* arow = agg  + (long long)(row0 + m) * DIM;
  const float* xrow = xdst + (long long)(row0 + m) * DIM;
  const float  scl  = invcnt[row0 + m];

  v8f acc0 = {}, acc1 = {}, acc2 = {}, acc3 = {};
#pragma unroll
  for (int kk = 0; kk < 16; ++kk) {
    const int k0 = kk * 4 + kb;
    v2f a;  a.x  = arow[k0] * scl;  a.y  = arow[k0 + 1] * scl;
    v2f ax; ax.x = xrow[k0];        ax.y = xrow[k0 + 1];
    v8f* accs[4] = {&acc0, &acc1, &acc2, &acc3};
#pragma unroll
    for (int nt = 0; nt < 4; ++nt) {
      v2f bl, br;
      bl.x = lwl[(k0    ) * DIM + nt * 16 + m];
      bl.y = lwl[(k0 + 1) * DIM + nt * 16 + m];
      br.x = lwr[(k0    ) * DIM + nt * 16 + m];
      br.y = lwr[(k0 + 1) * DIM + nt * 16 + m];
      *accs[nt] = __builtin_amdgcn_wmma_f32_16x16x4_f32(false, a,  false, bl,
                                                        (short)0, *accs[nt], false, false);
      *accs[nt] = __builtin_amdgcn_wmma_f32_16x16x4_f32(false, ax, false, br,
                                                        (short)0, *accs[nt], false, false);
    }
  }

  // store: C/D layout -> M = v + 8*half, N = nt*16 + (lane&15)
  v8f accs[4] = {acc0, acc1, acc2, acc3};
#pragma unroll
  for (int nt = 0; nt < 4; ++nt) {
    const int n = nt * 16 + m;
    const float bv = lb[n];
#pragma unroll
    for (int v = 0; v < 8; ++v) {
      float r = accs[nt][v] + bv;
      if (do_relu) r = fmaxf(r, 0.0f);
      out[(long long)(row0 + v + 8 * half) * DIM + n] = r;
    }
  }
}

// ---------------- WMMA decoder ----------------
// out[e] = sum_n (z_user[row[e]] @ Wu + bu)[n] * (z_rest[col[e]] @ Wr + br)[n]
// One wave handles 16 label edges (two gathered 16x64 @ 64x64 WMMA matmuls),
// then cross-lane shfl reduction over N inside each 16-lane half.
__global__ void decoder_wmma(const float* __restrict__ zu, const float* __restrict__ zr,
                             const int* __restrict__ rows, const int* __restrict__ cols,
                             const float* __restrict__ Wu, const float* __restrict__ bu,
                             const float* __restrict__ Wr, const float* __restrict__ br,
                             float* out, int nl) {
  __shared__ float lwu[DIM * DIM];
  __shared__ float lwr[DIM * DIM];
  __shared__ float lbu[DIM];
  __shared__ float lbr[DIM];
  for (int i = threadIdx.x; i < DIM * DIM; i += blockDim.x) {
    lwu[i] = Wu[i];
    lwr[i] = Wr[i];
  }
  if (threadIdx.x < DIM) { lbu[threadIdx.x] = bu[threadIdx.x]; lbr[threadIdx.x] = br[threadIdx.x]; }
  __syncthreads();

  const int lane = threadIdx.x & 31;
  const int wid  = threadIdx.x >> 5;
  const int e0   = (blockIdx.x * 8 + wid) * 16;
  if (e0 >= nl) return;

  const int m    = lane & 15;
  const int half = lane >> 4;
  const int kb   = half * 2;

  const int em   = (e0 + m < nl) ? (e0 + m) : (nl - 1);
  const float* urow = zu + (long long)rows[em] * DIM;
  const float* rrow = zr + (long long)cols[em] * DIM;

  v8f cu0 = {}, cu1 = {}, cu2 = {}, cu3 = {};
  v8f cr0 = {}, cr1 = {}, cr2 = {}, cr3 = {};
#pragma unroll
  for (int kk = 0; kk < 16; ++kk) {
    const int k0 = kk * 4 + kb;
    v2f au; au.x = urow[k0]; au.y = urow[k0 + 1];
    v2f ar; ar.x = rrow[k0]; ar.y = rrow[k0 + 1];
    v8f* pu[4] = {&cu0, &cu1, &cu2, &cu3};
    v8f* pr[4] = {&cr0, &cr1, &cr2, &cr3};
#pragma unroll
    for (int nt = 0; nt < 4; ++nt) {
      v2f b1, b2;
      b1.x = lwu[(k0    ) * DIM + nt * 16 + m];
      b1.y = lwu[(k0 + 1) * DIM + nt * 16 + m];
      b2.x = lwr[(k0    ) * DIM + nt * 16 + m];
      b2.y = lwr[(k0 + 1) * DIM + nt * 16 + m];
      *pu[nt] = __builtin_amdgcn_wmma_f32_16x16x4_f32(false, au, false, b1,
                                                      (short)0, *pu[nt], false, false);
      *pr[nt] = __builtin_amdgcn_wmma_f32_16x16x4_f32(false, ar, false, b2,
                                                      (short)0, *pr[nt], false, false);
    }
  }

  v8f cus[4] = {cu0, cu1, cu2, cu3};
  v8f crs[4] = {cr0, cr1, cr2, cr3};
#pragma unroll
  for (int v = 0; v < 8; ++v) {
    float p = 0.0f;
#pragma unroll
    for (int nt = 0; nt < 4; ++nt) {
      const int n = nt * 16 + m;
      p += (cus[nt][v] + lbu[n]) * (crs[nt][v] + lbr[n]);
    }
    // reduce over the 16 lanes of this half (N = 0..15 within each tile)
    p += __shfl_xor(p, 1, 32);
    p += __shfl_xor(p, 2, 32);
    p += __shfl_xor(p, 4, 32);
    p += __shfl_xor(p, 8, 32);
    const int eo = e0 + v + 8 * half;
    if (m == 0 && eo < nl) out[eo] = p;
  }
}

// ---------------- launcher ----------------
extern "C" void kernel_launch(void* const* d_in, const int* in_sizes, int n_in,
                              void* d_out, int out_size, void* d_ws, size_t ws_size,
                              hipStream_t stream) {
  const float* x_user  = (const float*)d_in[0];
  const float* x_rest  = (const float*)d_in[1];
  const float* W1l_u2r = (const float*)d_in[2];
  const float* b1_u2r  = (const float*)d_in[3];
  const float* W1r_u2r = (const float*)d_in[4];
  const float* W1l_r2u = (const float*)d_in[5];
  const float* b1_r2u  = (const float*)d_in[6];
  const float* W1r_r2u = (const float*)d_in[7];
  const float* W2l_u2r = (const float*)d_in[8];
  const float* b2_u2r  = (const float*)d_in[9];
  const float* W2r_u2r = (const float*)d_in[10];
  const float* W2l_r2u = (const float*)d_in[11];
  const float* b2_r2u  = (const float*)d_in[12];
  const float* W2r_r2u = (const float*)d_in[13];
  const float* Wu_dec  = (const float*)d_in[14];
  const float* bu_dec  = (const float*)d_in[15];
  const float* Wr_dec  = (const float*)d_in[16];
  const float* br_dec  = (const float*)d_in[17];
  const int*   e_u2r   = (const int*)d_in[18]; // [0..E)=user src, [E..2E)=rest dst
  const int*   eli     = (const int*)d_in[20]; // [0..L)=user row, [L..2L)=rest col

  float* ws = (float*)d_ws;
  float* agg_rest = ws + OFF_AGG_REST;
  float* agg_user = ws + OFF_AGG_USER;
  float* cnt_rest = ws + OFF_CNT_REST;
  float* cnt_user = ws + OFF_CNT_USER;
  float* h_rest   = ws + OFF_H_REST;
  float* h_user   = ws + OFF_H_USER;

  const int* su = e_u2r;           // user endpoint per edge
  const int* du = e_u2r + NEDGE;   // rest endpoint per edge

  const int TB = 256;
  const long long zn1 = (long long)OFF_H_REST; // aggs + counts region
  zero_f32<<<(unsigned)((zn1 + TB - 1) / TB), TB, 0, stream>>>(ws, zn1);

  count_edges<<<(NEDGE + TB - 1) / TB, TB, 0, stream>>>(su, du, cnt_rest, cnt_user, NEDGE);
  invert_cnt<<<((NREST + NUSER) + TB - 1) / TB, TB, 0, stream>>>(cnt_rest, NREST + NUSER);

  // ---- layer 1 ----
  const long long scat_threads = (long long)NEDGE * DIM;
  scatter_both<<<(unsigned)((scat_threads + TB - 1) / TB), TB, 0, stream>>>(
      x_user, x_rest, su, du, agg_rest, agg_user, NEDGE);

  sage_wmma<<<(NREST + 127) / 128, TB, 0, stream>>>(
      agg_rest, cnt_rest, x_rest, W1l_u2r, b1_u2r, W1r_u2r, h_rest, NREST, 1);
  sage_wmma<<<(NUSER + 127) / 128, TB, 0, stream>>>(
      agg_user, cnt_user, x_user, W1l_r2u, b1_r2u, W1r_r2u, h_user, NUSER, 1);

  // ---- layer 2 (re-zero aggs, scatter h, SAGE in place -> z lives in agg buffers) ----
  const long long zn2 = (long long)(NREST + NUSER) * DIM;
  zero_f32<<<(unsigned)((zn2 + TB - 1) / TB), TB, 0, stream>>>(ws, zn2);
  scatter_both<<<(unsigned)((scat_threads + TB - 1) / TB), TB, 0, stream>>>(
      h_user, h_rest, su, du, agg_rest, agg_user, NEDGE);

  sage_wmma<<<(NREST + 127) / 128, TB, 0, stream>>>(
      agg_rest, cnt_rest, h_rest, W2l_u2r, b2_u2r, W2r_u2r, agg_rest /*z_rest*/, NREST, 0);
  sage_wmma<<<(NUSER + 127) / 128, TB, 0, stream>>>(
      agg_user, cnt_user, h_user, W2l_r2u, b2_r2u, W2r_r2u, agg_user /*z_user*/, NUSER, 0);

  // ---- decoder ----
  decoder_wmma<<<(NLBL + 127) / 128, TB, 0, stream>>>(
      agg_user /*z_user*/, agg_rest /*z_rest*/, eli, eli + NLBL,
      Wu_dec, bu_dec, Wr_dec, br_dec, (float*)d_out, NLBL);

  (void)in_sizes; (void)n_in; (void)out_size; (void)ws_size;
}